// Ive_18571438588481
// MI455X (gfx1250) — compile-verified
//
#include <hip/hip_runtime.h>

// i1e(z) = exp(-|z|) * I1(z), elementwise, f32.
// Cephes single-precision coefficients (identical to XLA's i1e f32 expansion).

typedef __attribute__((ext_vector_type(4))) float v4f;

__device__ __forceinline__ float i1e_scalar(float x) {
    const float ax = __builtin_fabsf(x);

    // ---------- small regime: x <= 8 :  i1e = x * chbevl(x/2 - 2, A) ----------
    const float A[17] = {
         9.38153738649577178388E-9f, -4.44505912879632808065E-8f,
         2.00329475355213526229E-7f, -8.56872026469545474066E-7f,
         3.47025130813767847674E-6f, -1.32731636560394358279E-5f,
         4.78156510755005422638E-5f, -1.61760815825896745588E-4f,
         5.12285956168575772895E-4f, -1.51357245063125314899E-3f,
         4.15642294431288815669E-3f, -1.05640848946261981558E-2f,
         2.47264490306265168283E-2f, -5.29459812080949914269E-2f,
         1.02643658689847095384E-1f, -1.76416518357834055153E-1f,
         2.52587186443633654823E-1f};

    float ys = __builtin_fmaf(ax, 0.5f, -2.0f);
    float sb0 = 0.0f, sb1 = 0.0f, sb2 = 0.0f;
#pragma unroll
    for (int i = 0; i < 17; ++i) {
        sb2 = sb1;
        sb1 = sb0;
        sb0 = __builtin_fmaf(ys, sb1, A[i]) - sb2;   // Chebyshev recurrence
    }
    const float small_v = 0.5f * (sb0 - sb2) * ax;

    // ---------- large regime: x > 8 :  i1e = chbevl(32/x - 2, B) / sqrt(x) ----------
    const float B[7] = {
        -3.83538038596423702205E-9f, -2.63146884688951950684E-8f,
        -2.51223623787020892529E-7f, -3.88256480887769039346E-6f,
        -1.10588938762623716291E-4f, -9.76109749136146840777E-3f,
         7.78576235018280120474E-1f};

    // v_rcp_f32 (~1 ulp) instead of full IEEE division expansion.
    const float yl = __builtin_fmaf(32.0f, __builtin_amdgcn_rcpf(ax), -2.0f);
    float lb0 = 0.0f, lb1 = 0.0f, lb2 = 0.0f;
#pragma unroll
    for (int i = 0; i < 7; ++i) {
        lb2 = lb1;
        lb1 = lb0;
        lb0 = __builtin_fmaf(yl, lb1, B[i]) - lb2;
    }
    // v_rsq_f32 for 1/sqrt(x).
    const float large_v = 0.5f * (lb0 - lb2) * __builtin_amdgcn_rsqf(ax);

    const float v = (ax <= 8.0f) ? small_v : large_v;
    return __builtin_copysignf(v, x);   // i1e is odd
}

__device__ __forceinline__ v4f i1e_v4(v4f v) {
    v4f r;
    r.x = i1e_scalar(v.x);
    r.y = i1e_scalar(v.y);
    r.z = i1e_scalar(v.z);
    r.w = i1e_scalar(v.w);
    return r;
}

// 8 elements per thread: amortizes SALU coefficient setup / ID math over 2x
// the data, and gives 8 independent Chebyshev chains -> 4 v_pk_fma_f32 streams.
__global__ void __launch_bounds__(256)
i1e_v8_kernel(const float* __restrict__ in, float* __restrict__ out, int n8) {
    const int t = blockIdx.x * 256 + threadIdx.x;
    if (t >= n8) return;

    const v4f* inq = reinterpret_cast<const v4f*>(in);
    v4f*       outq = reinterpret_cast<v4f*>(out);
    const int  q = t * 2;

    // Two 16-byte non-temporal streaming loads (single-touch streams;
    // don't thrash the 192 MB L2).
    v4f a = __builtin_nontemporal_load(inq + q);
    v4f b = __builtin_nontemporal_load(inq + q + 1);

    v4f ra = i1e_v4(a);
    v4f rb = i1e_v4(b);

    __builtin_nontemporal_store(ra, outq + q);
    __builtin_nontemporal_store(rb, outq + q + 1);
}

__global__ void __launch_bounds__(64)
i1e_tail_kernel(const float* __restrict__ in, float* __restrict__ out,
                int start, int n) {
    const int i = start + blockIdx.x * 64 + threadIdx.x;
    if (i < n) out[i] = i1e_scalar(in[i]);
}

extern "C" void kernel_launch(void* const* d_in, const int* in_sizes, int n_in,
                              void* d_out, int out_size, void* d_ws, size_t ws_size,
                              hipStream_t stream) {
    (void)in_sizes; (void)n_in; (void)d_ws; (void)ws_size;

    const float* z = reinterpret_cast<const float*>(d_in[0]);
    float* out     = reinterpret_cast<float*>(d_out);

    const int n  = out_size;        // 16384*8192 = 134,217,728
    const int n8 = n >> 3;          // 8-element groups (divisible here)

    if (n8 > 0) {
        const int threads = 256;
        const int blocks  = (n8 + threads - 1) / threads;
        i1e_v8_kernel<<<blocks, threads, 0, stream>>>(z, out, n8);
    }
    const int rem = n & 7;
    if (rem > 0) {
        i1e_tail_kernel<<<1, 64, 0, stream>>>(z, out, n8 << 3, n);
    }
}